// NTXentLoss_3204045603652
// MI455X (gfx1250) — compile-verified
//
#include <hip/hip_runtime.h>
#include <hip/hip_bf16.h>

// NT-Xent loss, fused normalize -> WMMA fp32 GEMM -> exp/row-sum -> logsumexp.
// n = 4096 rows, d = 512, T = 0.5 (inv = 2.0), eps = 1e-8.

#define NT_B 2048
#define NT_N 4096
#define NT_D 512

typedef __attribute__((ext_vector_type(2))) float v2f;
typedef __attribute__((ext_vector_type(8))) float v8f;

// ---------------------------------------------------------------------------
// Kernel 1: zn[row] = z[row] / max(||z[row]||, eps), z = concat(z_i, z_j)
// grid = 4096 blocks, 256 threads (8 wave32), each thread owns 2 elements.
// ---------------------------------------------------------------------------
__global__ __launch_bounds__(256) void nt_normalize(const float* __restrict__ zi,
                                                    const float* __restrict__ zj,
                                                    float* __restrict__ zn) {
  const int row = blockIdx.x;
  const float* src = (row < NT_B) ? (zi + (size_t)row * NT_D)
                                  : (zj + (size_t)(row - NT_B) * NT_D);
  const float2* s2 = (const float2*)src;
  const float2 v = s2[threadIdx.x];
  float ss = v.x * v.x + v.y * v.y;
#pragma unroll
  for (int off = 16; off > 0; off >>= 1) ss += __shfl_down(ss, off, 32);

  __shared__ float red[8];
  __shared__ float sinv;
  if ((threadIdx.x & 31) == 0) red[threadIdx.x >> 5] = ss;
  __syncthreads();
  if (threadIdx.x == 0) {
    float t = 0.0f;
#pragma unroll
    for (int i = 0; i < 8; ++i) t += red[i];
    sinv = 1.0f / fmaxf(sqrtf(t), 1e-8f);
  }
  __syncthreads();
  const float inv = sinv;
  float2* o2 = (float2*)(zn + (size_t)row * NT_D);
  float2 w;
  w.x = v.x * inv;
  w.y = v.y * inv;
  o2[threadIdx.x] = w;
}

// ---------------------------------------------------------------------------
// Kernel 2: for a 16-row stripe, sweep all 4096 columns with
// V_WMMA_F32_16X16X4_F32 tiles (K=512 -> 128 wmma per 16x16 tile).
// Each wave carries TWO adjacent column tiles (two v8f accumulators) so each
// ds_load'd A fragment feeds two wmmas. Fused epilogue accumulates
//   rowsum[i] = sum_{j != i} exp(2 * S_ij)     (diagonal skipped in-register)
//   pos[i]    = S_{i,(i+B) mod n}              (branchless select-accumulate)
// Block = 128 threads = 4 wave32; A stripe in LDS (padded stride 516 floats
// -> the 32-lane ds_load_b64 A-fragment hits 32 distinct banks).
// grid = 256 blocks.
// ---------------------------------------------------------------------------
__global__ __launch_bounds__(128) void nt_sim_tiles(const float* __restrict__ zn,
                                                    float* __restrict__ rowsum,
                                                    float* __restrict__ pos) {
  __shared__ float ldsA[16][516];
  __shared__ float lsum[16];
  __shared__ float lpos[16];

  const int tid = threadIdx.x;
  const int rowBase = blockIdx.x * 16;

  // Cooperative fill of the 16x512 A stripe (16 float4 per thread).
  {
    const int r  = tid >> 3;          // 0..15
    const int c0 = (tid & 7) * 64;    // 8 threads x 64 floats per row
    const float4* g4 = (const float4*)(zn + (size_t)(rowBase + r) * NT_D + c0);
#pragma unroll
    for (int i = 0; i < 16; ++i) {
      const float4 v = g4[i];
      const int c = c0 + i * 4;
      ldsA[r][c + 0] = v.x;
      ldsA[r][c + 1] = v.y;
      ldsA[r][c + 2] = v.z;
      ldsA[r][c + 3] = v.w;
    }
  }
  if (tid < 16) { lsum[tid] = 0.0f; lpos[tid] = 0.0f; }
  __syncthreads();

  const int lane = tid & 31;
  const int wave = tid >> 5;
  const int m  = lane & 15;   // A: M row / B,C: N column within tile
  const int hi = lane >> 4;   // lane-half selects K pair (A/B) or M+8 (C)

  float esum[8];
  float pval[8];
#pragma unroll
  for (int r = 0; r < 8; ++r) { esum[r] = 0.0f; pval[r] = 0.0f; }

  // Each wave owns pairs of adjacent 16-column tiles; 32 pair-iterations.
  for (int j = wave; j < NT_N / 32; j += 4) {
    const int ct0  = (2 * j) * 16;
    const int col0 = ct0 + m;
    const int col1 = col0 + 16;
    const float2* bp0 = (const float2*)(zn + (size_t)col0 * NT_D);
    const float2* bp1 = (const float2*)(zn + (size_t)col1 * NT_D);

    v8f c0 = {};
    v8f c1 = {};
#pragma unroll 8
    for (int k0 = 0; k0 < NT_D; k0 += 4) {
      // A fragment: lane(m,hi) holds zn[rowBase+m][k0+2*hi .. +1] (from LDS)
      const float2 av = *(const float2*)&ldsA[m][k0 + 2 * hi];
      // B fragments (S symmetric => same per-lane addressing off the column row)
      const float2 bv0 = bp0[(k0 >> 1) + hi];
      const float2 bv1 = bp1[(k0 >> 1) + hi];
      v2f a;  a.x  = av.x;  a.y  = av.y;
      v2f b0; b0.x = bv0.x; b0.y = bv0.y;
      v2f b1; b1.x = bv1.x; b1.y = bv1.y;
      c0 = __builtin_amdgcn_wmma_f32_16x16x4_f32(
          false, a, false, b0, (short)0, c0, false, false);
      c1 = __builtin_amdgcn_wmma_f32_16x16x4_f32(
          false, a, false, b1, (short)0, c1, false, false);
    }

    // C fragment layout: VGPR r, this lane -> (row = rowBase + r + 8*hi, col)
#pragma unroll
    for (int r = 0; r < 8; ++r) {
      const int row = rowBase + r + 8 * hi;
      const int partner = (row + NT_B) & (NT_N - 1);
      const float s0 = c0[r];
      const float s1 = c1[r];
      pval[r] += (col0 == partner) ? s0 : 0.0f;   // branchless pos capture
      pval[r] += (col1 == partner) ? s1 : 0.0f;
      esum[r] += (col0 == row) ? 0.0f : __expf(s0 * 2.0f);   // skip diagonal
      esum[r] += (col1 == row) ? 0.0f : __expf(s1 * 2.0f);
    }
  }

  // Reduce per-row partials across lanes/waves via LDS float atomics
  // (pval contributes exact 0.0 from every lane except the partner owner).
#pragma unroll
  for (int r = 0; r < 8; ++r) {
    atomicAdd(&lsum[r + 8 * hi], esum[r]);
    atomicAdd(&lpos[r + 8 * hi], pval[r]);
  }
  __syncthreads();
  if (tid < 16) {
    rowsum[rowBase + tid] = lsum[tid];
    pos[rowBase + tid]    = lpos[tid];
  }
}

// ---------------------------------------------------------------------------
// Kernel 3: loss = mean_i( log( exp(2*pos_i) + rowsum_i ) - 2*pos_i )
// Single block of 1024 threads (32 wave32).
// ---------------------------------------------------------------------------
__global__ __launch_bounds__(1024) void nt_final(const float* __restrict__ rowsum,
                                                 const float* __restrict__ pos,
                                                 float* __restrict__ out) {
  float acc = 0.0f;
  for (int i = threadIdx.x; i < NT_N; i += 1024) {
    const float p = pos[i] * 2.0f;                 // pos / T
    acc += __logf(rowsum[i] + __expf(p)) - p;      // logsumexp - logits[0]
  }
#pragma unroll
  for (int off = 16; off > 0; off >>= 1) acc += __shfl_down(acc, off, 32);
  __shared__ float red[32];
  if ((threadIdx.x & 31) == 0) red[threadIdx.x >> 5] = acc;
  __syncthreads();
  if (threadIdx.x == 0) {
    float t = 0.0f;
#pragma unroll
    for (int i = 0; i < 32; ++i) t += red[i];
    out[0] = t * (1.0f / (float)NT_N);
  }
}

// ---------------------------------------------------------------------------
extern "C" void kernel_launch(void* const* d_in, const int* in_sizes, int n_in,
                              void* d_out, int out_size, void* d_ws, size_t ws_size,
                              hipStream_t stream) {
  (void)in_sizes; (void)n_in; (void)out_size; (void)ws_size;

  const float* zi = (const float*)d_in[0];
  const float* zj = (const float*)d_in[1];
  float* out = (float*)d_out;

  char* ws = (char*)d_ws;
  float* zn     = (float*)ws;                                         // 4096*512 f32 (8 MB)
  float* rowsum = (float*)(ws + (size_t)NT_N * NT_D * sizeof(float)); // 4096 f32
  float* pos    = rowsum + NT_N;                                      // 4096 f32

  nt_normalize<<<NT_N, 256, 0, stream>>>(zi, zj, zn);
  nt_sim_tiles<<<NT_N / 16, 128, 0, stream>>>(zn, rowsum, pos);
  nt_final<<<1, 1024, 0, stream>>>(rowsum, pos, out);
}